// SvdLSTMCell_79405355369081
// MI455X (gfx1250) — compile-verified
//
#include <hip/hip_runtime.h>
#include <math.h>

#define B_ 128
#define I_ 64
#define H_ 384

typedef float v2f __attribute__((ext_vector_type(2)));
typedef float v8f __attribute__((ext_vector_type(8)));
typedef unsigned int v4u __attribute__((ext_vector_type(4)));
typedef int v8i __attribute__((ext_vector_type(8)));
typedef int v4i __attribute__((ext_vector_type(4)));

__device__ __forceinline__ float sigf(float z) { return 1.0f / (1.0f + __expf(-z)); }

// ---------------------------------------------------------------------------
// Tensor Data Mover: 2-D tile load (Global -> LDS), D# per CDNA5 ISA §8.3/8.4.
// 6-arg builtin form (this toolchain): (g0, g1, g2, g3, spare, cpol).
// ---------------------------------------------------------------------------
#if __has_builtin(__builtin_amdgcn_tensor_load_to_lds)
#define HAVE_TDM 1
__device__ __forceinline__ void tdm_load_2d(void* lds_ptr, const float* gptr,
                                            unsigned flags, int tile_d0,
                                            int tile_d1, int stride) {
    unsigned long long ga = (unsigned long long)(uintptr_t)gptr;
    v4u g0;
    g0[0] = 1u;                                             // count=1 (valid D#)
    g0[1] = (unsigned)(uintptr_t)lds_ptr;                   // LDS byte address
    g0[2] = (unsigned)ga;                                   // global addr [31:0]
    g0[3] = (unsigned)((ga >> 32) & 0x01ffffffu) | 0x80000000u; // addr[56:32] | type=2
    v8i g1;
    g1[0] = (int)flags;                                     // data_size=4B [+pad cfg]
    g1[1] = (int)((unsigned)(H_ & 0xffff) << 16);           // tensor_dim0 lo16
    g1[2] = (int)(((unsigned)(H_ >> 16) & 0xffffu) |
                  ((unsigned)(H_ & 0xffff) << 16));         // dim0 hi16 | dim1 lo16
    g1[3] = (int)(((unsigned)(H_ >> 16) & 0xffffu) |
                  ((unsigned)tile_d0 << 16));               // dim1 hi16 | tile_dim0
    g1[4] = tile_d1 & 0xffff;                               // tile_dim1 (tile_dim2=0)
    g1[5] = stride;                                         // tensor_dim0_stride lo32
    g1[6] = 0;
    g1[7] = 0;
    v4i z4 = {0, 0, 0, 0};
    v8i z8 = {0, 0, 0, 0, 0, 0, 0, 0};
    __builtin_amdgcn_tensor_load_to_lds(g0, g1, z4, z4, z8, 0);
}
#else
#define HAVE_TDM 0
#endif

#define TDM_FLAGS_PAD8 (0x00020000u | (1u << 20) | (6u << 22) | (7u << 25))

// ---------------------------------------------------------------------------
// 1) cx -> column-major Acm; Vcm = I.
// ---------------------------------------------------------------------------
__global__ __launch_bounds__(256)
void prep_kernel(const float* __restrict__ cx, float* __restrict__ Acm,
                 float* __restrict__ Vcm) {
    __shared__ float tile[32][33];
    const int b = blockIdx.z;
    const int r0 = blockIdx.y * 32, c0 = blockIdx.x * 32;
    const float* src = cx + (size_t)b * H_ * H_;
    for (int rr = threadIdx.y; rr < 32; rr += blockDim.y)
        tile[rr][threadIdx.x] = src[(size_t)(r0 + rr) * H_ + c0 + threadIdx.x];
    __syncthreads();
    float* Ab = Acm + (size_t)b * H_ * H_;
    float* Vb = Vcm + (size_t)b * H_ * H_;
    for (int cc = threadIdx.y; cc < 32; cc += blockDim.y) {
        const int col = c0 + cc, row = r0 + threadIdx.x;
        Ab[(size_t)col * H_ + row] = tile[threadIdx.x][cc];
        Vb[(size_t)col * H_ + row] = (col == row) ? 1.0f : 0.0f;
    }
}

// ---------------------------------------------------------------------------
// 2) Batched one-sided Jacobi (Hestenes). 8 wave32 per batch matrix, 192
//    disjoint pairs/round (24/wave). Lanes own 2 consecutive elements so all
//    column traffic is coalesced b64; wave32 shuffle reductions.
// ---------------------------------------------------------------------------
template <bool ACCV>
__global__ __launch_bounds__(256)
void jacobi_kernel(float* __restrict__ A, float* __restrict__ V, int nsweeps) {
    const int b = blockIdx.x;
    float* Ab = A + (size_t)b * H_ * H_;
    float* Vb = ACCV ? (V + (size_t)b * H_ * H_) : nullptr;
    const int lane = threadIdx.x & 31;
    const int wave = threadIdx.x >> 5;
    const int nw = blockDim.x >> 5;
    const int n1 = H_ - 1;
    const int e0 = 2 * lane;
    for (int sw = 0; sw < nsweeps; ++sw) {
        for (int r = 0; r < n1; ++r) {
            for (int k = wave; k < H_ / 2; k += nw) {
                int p, q;
                if (k == 0) { p = r % n1; q = n1; }
                else { p = (r + k) % n1; q = (r - k + 2 * n1) % n1; }
                if (p > q) { int t = p; p = q; q = t; }
                float* cp = Ab + (size_t)p * H_;
                float* cq = Ab + (size_t)q * H_;
                float app = 0.f, aqq = 0.f, apq = 0.f;
                #pragma unroll
                for (int i0 = 0; i0 < H_; i0 += 64) {
                    const v2f a = *(const v2f*)&cp[i0 + e0];
                    const v2f d = *(const v2f*)&cq[i0 + e0];
                    app = fmaf(a[0], a[0], fmaf(a[1], a[1], app));
                    aqq = fmaf(d[0], d[0], fmaf(d[1], d[1], aqq));
                    apq = fmaf(a[0], d[0], fmaf(a[1], d[1], apq));
                }
                #pragma unroll
                for (int m = 16; m >= 1; m >>= 1) {
                    app += __shfl_xor(app, m, 32);
                    aqq += __shfl_xor(aqq, m, 32);
                    apq += __shfl_xor(apq, m, 32);
                }
                if (fabsf(apq) > 1e-12f) {
                    const float tau = (aqq - app) / (2.0f * apq);
                    const float t = copysignf(1.0f, tau) /
                                    (fabsf(tau) + sqrtf(fmaf(tau, tau, 1.0f)));
                    const float c = rsqrtf(fmaf(t, t, 1.0f));
                    const float s = t * c;
                    #pragma unroll
                    for (int i0 = 0; i0 < H_; i0 += 64) {
                        v2f a = *(v2f*)&cp[i0 + e0];
                        v2f d = *(v2f*)&cq[i0 + e0];
                        v2f na, nd;
                        na[0] = fmaf(c, a[0], -s * d[0]);
                        na[1] = fmaf(c, a[1], -s * d[1]);
                        nd[0] = fmaf(s, a[0],  c * d[0]);
                        nd[1] = fmaf(s, a[1],  c * d[1]);
                        *(v2f*)&cp[i0 + e0] = na;
                        *(v2f*)&cq[i0 + e0] = nd;
                        if (ACCV) {
                            float* vp = Vb + (size_t)p * H_;
                            float* vq = Vb + (size_t)q * H_;
                            v2f va = *(v2f*)&vp[i0 + e0];
                            v2f vd = *(v2f*)&vq[i0 + e0];
                            v2f nva, nvd;
                            nva[0] = fmaf(c, va[0], -s * vd[0]);
                            nva[1] = fmaf(c, va[1], -s * vd[1]);
                            nvd[0] = fmaf(s, va[0],  c * vd[0]);
                            nvd[1] = fmaf(s, va[1],  c * vd[1]);
                            *(v2f*)&vp[i0 + e0] = nva;
                            *(v2f*)&vq[i0 + e0] = nvd;
                        }
                    }
                }
            }
            __syncthreads();
        }
    }
}

// ---------------------------------------------------------------------------
// 3) Singular-value ranking: scl[b][j] = hx[b][rank_j] / s_j
// ---------------------------------------------------------------------------
__global__ __launch_bounds__(H_)
void rank_scale_kernel(const float* __restrict__ Acm, const float* __restrict__ hx,
                       float* __restrict__ scl) {
    __shared__ float s[H_];
    const int b = blockIdx.x, j = threadIdx.x;
    const float* col = Acm + (size_t)b * H_ * H_ + (size_t)j * H_;
    float nn = 0.f;
    for (int r = 0; r < H_; ++r) { const float a = col[r]; nn = fmaf(a, a, nn); }
    const float sj = sqrtf(nn);
    s[j] = sj;
    __syncthreads();
    int rank = 0;
    for (int k = 0; k < H_; ++k) {
        const float sk = s[k];
        rank += (sk > sj) || (sk == sj && k < j);
    }
    scl[(size_t)b * H_ + j] = hx[(size_t)b * H_ + rank] / fmaxf(sj, 1e-20f);
}

// ---------------------------------------------------------------------------
// 4) Fused WMMA kernel, 128x16 output panel per 8-wave block.
//    A panel: TDM, double-buffered (DMA c+1 overlaps compute c), LDS-padded.
//    V panel: cooperative transposed copy with scale folded -> ds_load_b64 B.
//    Operands batched 8 WMMA steps per wait.
// ---------------------------------------------------------------------------
#define KC 64
#define NCH (H_ / KC)
#define APAD 136
#define VPAD 68

__global__ __launch_bounds__(256)
void fused_gates_kernel(const float* __restrict__ x, const float* __restrict__ cx,
                        const float* __restrict__ Wi, const float* __restrict__ Wit,
                        const float* __restrict__ bi,
                        const float* __restrict__ Wf, const float* __restrict__ Wft,
                        const float* __restrict__ bf,
                        const float* __restrict__ Wc, const float* __restrict__ Wct,
                        const float* __restrict__ bc,
                        const float* __restrict__ Acm, const float* __restrict__ Vcm,
                        const float* __restrict__ scl,
                        float* __restrict__ Cnew) {
    __shared__ float Abuf[2][KC * APAD];
    __shared__ float VbufT[2][16 * VPAD];

    const int lane = threadIdx.x;
    const int lo = lane & 15;
    const int hi = lane >> 4;
    const int ty = threadIdx.y;
    const int tid = ty * 32 + lane;
    const int b = blockIdx.z;
    const int upan = blockIdx.x * 128;
    const int u0 = upan + ty * 16;
    const int v0 = blockIdx.y * 16;

    const float* Ab = Acm + (size_t)b * H_ * H_;
    const float* Vb = Vcm + (size_t)b * H_ * H_;
    const float* sc = scl + (size_t)b * H_;
    const float* xb = x + (size_t)b * I_;

    v8f ch = {}; v8f ci = {}; v8f cf = {}; v8f cg = {};

#if HAVE_TDM
    if (ty == 0)
        tdm_load_2d(&Abuf[0][0], Ab + upan, TDM_FLAGS_PAD8, 128, KC, H_);
#endif
    for (int c = 0; c < NCH; ++c) {
        const int k0 = c * KC;
        float* Acur = &Abuf[c & 1][0];
        float* Vcur = &VbufT[c & 1][0];
#if HAVE_TDM
        if (ty == 0) __builtin_amdgcn_s_wait_tensorcnt(0);
        __syncthreads();
        if (ty == 0 && c + 1 < NCH)
            tdm_load_2d(&Abuf[(c + 1) & 1][0],
                        Ab + (size_t)(k0 + KC) * H_ + upan,
                        TDM_FLAGS_PAD8, 128, KC, H_);
#else
        __syncthreads();
        for (int idx = tid; idx < KC * 128; idx += 256)
            Acur[(idx >> 7) * APAD + (idx & 127)] =
                Ab[(size_t)(k0 + (idx >> 7)) * H_ + upan + (idx & 127)];
#endif
        for (int idx = tid; idx < KC * 16; idx += 256) {
            const int k = idx >> 4, n = idx & 15;
            Vcur[n * VPAD + k] = Vb[(size_t)(k0 + k) * H_ + v0 + n] * sc[k0 + k];
        }
        __syncthreads();

        #pragma unroll
        for (int half = 0; half < 2; ++half) {
            v2f aop[8], bop[8];
            #pragma unroll
            for (int j = 0; j < 8; ++j) {
                const int ka = half * 32 + j * 4 + 2 * hi;
                aop[j][0] = Acur[ka * APAD + ty * 16 + lo];
                aop[j][1] = Acur[(ka + 1) * APAD + ty * 16 + lo];
                bop[j] = *(const v2f*)&Vcur[lo * VPAD + ka];
            }
            #pragma unroll
            for (int j = 0; j < 8; ++j)
                ch = __builtin_amdgcn_wmma_f32_16x16x4_f32(
                        false, aop[j], false, bop[j], (short)0, ch, false, false);
        }
    }

    #pragma unroll 4
    for (int k0 = 0; k0 < I_; k0 += 4) {
        const int ka = k0 + 2 * hi;
        const v2f xv = *(const v2f*)&xb[ka];
        v2f av, bv;
        const v2f wi = *(const v2f*)&Wi[(size_t)(u0 + lo) * I_ + ka];
        av[0] = wi[0] * xv[0]; av[1] = wi[1] * xv[1];
        bv = *(const v2f*)&Wit[(size_t)(v0 + lo) * I_ + ka];
        ci = __builtin_amdgcn_wmma_f32_16x16x4_f32(false, av, false, bv,
                                                   (short)0, ci, false, false);
        const v2f wf = *(const v2f*)&Wf[(size_t)(u0 + lo) * I_ + ka];
        av[0] = wf[0] * xv[0]; av[1] = wf[1] * xv[1];
        bv = *(const v2f*)&Wft[(size_t)(v0 + lo) * I_ + ka];
        cf = __builtin_amdgcn_wmma_f32_16x16x4_f32(false, av, false, bv,
                                                   (short)0, cf, false, false);
        const v2f wc = *(const v2f*)&Wc[(size_t)(u0 + lo) * I_ + ka];
        av[0] = wc[0] * xv[0]; av[1] = wc[1] * xv[1];
        bv = *(const v2f*)&Wct[(size_t)(v0 + lo) * I_ + ka];
        cg = __builtin_amdgcn_wmma_f32_16x16x4_f32(false, av, false, bv,
                                                   (short)0, cg, false, false);
    }

    #pragma unroll
    for (int e = 0; e < 8; ++e) {
        const int u = u0 + e + 8 * hi;
        const int v = v0 + lo;
        const size_t gidx = (size_t)u * H_ + v;
        const size_t cidx = (size_t)b * H_ * H_ + gidx;
        const float hm = ch[e];
        const float iv = sigf(ci[e] + hm + bi[gidx]);
        const float fv = sigf(cf[e] + hm + bf[gidx]);
        const float gv = tanhf(cg[e] + hm + bc[gidx]);
        Cnew[cidx] = fmaf(fv, cx[cidx], iv * gv);
    }
}

// ---------------------------------------------------------------------------
// 5) Output gate: one wave32 per (b,h) row; coalesced reads + shuffle reduce.
// ---------------------------------------------------------------------------
__global__ __launch_bounds__(256)
void out_gate_kernel(const float* __restrict__ x, const float* __restrict__ hx,
                     const float* __restrict__ Wo, const float* __restrict__ Who,
                     const float* __restrict__ bo, float* __restrict__ o) {
    const int lane = threadIdx.x & 31;
    const int row = blockIdx.x * 8 + (threadIdx.x >> 5);
    if (row >= B_ * H_) return;
    const int b = row / H_, h = row % H_;
    float acc = 0.f;
    #pragma unroll
    for (int i = lane; i < I_; i += 32)
        acc = fmaf(Wo[(size_t)h * I_ + i], x[(size_t)b * I_ + i], acc);
    #pragma unroll 4
    for (int k = lane; k < H_; k += 32)
        acc = fmaf(Who[(size_t)h * H_ + k], hx[(size_t)b * H_ + k], acc);
    #pragma unroll
    for (int m = 16; m >= 1; m >>= 1) acc += __shfl_xor(acc, m, 32);
    if (lane == 0) o[row] = sigf(acc + bo[h]);
}

// ---------------------------------------------------------------------------
// 6) Final: sigma(cx_new), ranked descending, hx_new = o * sigmoid(S).
// ---------------------------------------------------------------------------
__global__ __launch_bounds__(H_)
void final_kernel(const float* __restrict__ Crow, const float* __restrict__ o,
                  float* __restrict__ out) {
    __shared__ float s[H_];
    const int b = blockIdx.x, j = threadIdx.x;
    const float* col = Crow + (size_t)b * H_ * H_ + (size_t)j * H_;
    float nn = 0.f;
    for (int r = 0; r < H_; ++r) { const float a = col[r]; nn = fmaf(a, a, nn); }
    const float sj = sqrtf(nn);
    s[j] = sj;
    __syncthreads();
    int rank = 0;
    for (int k = 0; k < H_; ++k) {
        const float sk = s[k];
        rank += (sk > sj) || (sk == sj && k < j);
    }
    out[(size_t)b * H_ + rank] = o[(size_t)b * H_ + rank] * sigf(sj);
}

// ---------------------------------------------------------------------------
extern "C" void kernel_launch(void* const* d_in, const int* in_sizes, int n_in,
                              void* d_out, int out_size, void* d_ws, size_t ws_size,
                              hipStream_t stream) {
    (void)in_sizes; (void)n_in; (void)out_size; (void)ws_size;
    const float* x   = (const float*)d_in[0];
    const float* hx  = (const float*)d_in[1];
    const float* cx  = (const float*)d_in[2];
    const float* Wi  = (const float*)d_in[3];
    const float* Wit = (const float*)d_in[4];
    const float* bi  = (const float*)d_in[5];
    const float* Wf  = (const float*)d_in[6];
    const float* Wft = (const float*)d_in[7];
    const float* bf  = (const float*)d_in[8];
    const float* Wc  = (const float*)d_in[9];
    const float* Wct = (const float*)d_in[10];
    const float* bc  = (const float*)d_in[11];
    const float* Wo  = (const float*)d_in[12];
    const float* Who = (const float*)d_in[13];
    const float* bo  = (const float*)d_in[14];
    float* out = (float*)d_out;

    float* ws = (float*)d_ws;
    const size_t mat = (size_t)B_ * H_ * H_;
    float* Acm  = ws;
    float* Vcm  = Acm + mat;
    float* Cnew = Vcm + mat;
    float* scl  = Cnew + mat;
    float* ovec = scl + (size_t)B_ * H_;

    const int NSWEEPS = 8;
    dim3 tb(32, 8);

    prep_kernel<<<dim3(H_ / 32, H_ / 32, B_), tb, 0, stream>>>(cx, Acm, Vcm);
    jacobi_kernel<true><<<dim3(B_), dim3(256), 0, stream>>>(Acm, Vcm, NSWEEPS);
    rank_scale_kernel<<<dim3(B_), dim3(H_), 0, stream>>>(Acm, hx, scl);
    fused_gates_kernel<<<dim3(H_ / 128, H_ / 16, B_), tb, 0, stream>>>(
        x, cx, Wi, Wit, bi, Wf, Wft, bf, Wc, Wct, bc, Acm, Vcm, scl, Cnew);
    out_gate_kernel<<<dim3(B_ * H_ / 8), dim3(256), 0, stream>>>(
        x, hx, Wo, Who, bo, ovec);
    jacobi_kernel<false><<<dim3(B_), dim3(256), 0, stream>>>(Cnew, nullptr, NSWEEPS);
    final_kernel<<<dim3(B_), dim3(H_), 0, stream>>>(Cnew, ovec, out);
}